// PointNet2_4355096838383
// MI455X (gfx1250) — compile-verified
//
#include <hip/hip_runtime.h>
#include <stdint.h>

#define B_   16
#define N_   131072
#define NP1  1024
#define OUTW 336   // 256 + 64 + 16 per batch

// Non-contracted distance: matches XLA's separate mul/add HLOs (no FMA fusion).
__device__ __forceinline__ float dist3(float dx, float dy, float dz) {
    return __fadd_rn(__fadd_rn(__fmul_rn(dx, dx), __fmul_rn(dy, dy)),
                     __fmul_rn(dz, dz));
}

// ---- (max value, min index) semilattice combine: matches jnp.argmax tie-break ----
__device__ __forceinline__ void red2(float& v, int& i, float ov, int oi) {
    if (ov > v || (ov == v && oi < i)) { v = ov; i = oi; }
}

__device__ __forceinline__ void waveReduce(float& v, int& i) {
#pragma unroll
    for (int off = 16; off > 0; off >>= 1) {
        float ov = __shfl_xor(v, off, 32);
        int   oi = __shfl_xor(i, off, 32);
        red2(v, i, ov, oi);
    }
}

// Combine that also carries the winning point's coordinates (pivot for next step).
__device__ __forceinline__ void red5(float& v, int& i, float& x, float& y, float& z,
                                     float ov, int oi, float ox, float oy, float oz) {
    if (ov > v || (ov == v && oi < i)) { v = ov; i = oi; x = ox; y = oy; z = oz; }
}

__device__ __forceinline__ void waveReduce5(float& v, int& i, float& x, float& y, float& z) {
#pragma unroll
    for (int off = 16; off > 0; off >>= 1) {
        float ov = __shfl_xor(v, off, 32);
        int   oi = __shfl_xor(i, off, 32);
        float ox = __shfl_xor(x, off, 32);
        float oy = __shfl_xor(y, off, 32);
        float oz = __shfl_xor(z, off, 32);
        red5(v, i, x, y, z, ov, oi, ox, oy, oz);
    }
}

// ---- AoS (B,N,3) -> SoA xs/ys/zs, init dists to 1e10 ----
__global__ void prep_kernel(const float* __restrict__ x,
                            float* __restrict__ xs, float* __restrict__ ys,
                            float* __restrict__ zs, float* __restrict__ dd) {
    size_t i = (size_t)blockIdx.x * blockDim.x + threadIdx.x;
    if (i < (size_t)B_ * N_) {
        const float* p = x + i * 3;
        xs[i] = p[0]; ys[i] = p[1]; zs[i] = p[2];
        dd[i] = 1e10f;
    }
}

// ---- Stage 1: FPS 131072 -> 1024, one workgroup (32 wave32) per batch.
//      Pivot coords travel through the reduction: no dependent global load
//      on the serial critical path. ----
__global__ __launch_bounds__(1024)
void fps_big(const float4* xs4, const float4* ys4, const float4* zs4, float4* dd4,
             const float* xs, const float* ys, const float* zs,
             int* __restrict__ idx1) {
    const int b    = blockIdx.x;
    const int tid  = threadIdx.x;
    const int lane = tid & 31;
    const int wid  = tid >> 5;

    __shared__ float sv[32], spx[32], spy[32], spz[32];
    __shared__ int   si[32];
    __shared__ float sP[3];

    const size_t base4 = (size_t)b * (N_ / 4);
    const size_t base  = (size_t)b * N_;

    if (tid == 0) idx1[b * NP1] = 0;   // first sampled index is 0 (mmcv convention)

    // Initial pivot = point 0 (broadcast load, once, off the critical loop).
    float px = xs[base], py = ys[base], pz = zs[base];

    for (int s = 1; s < NP1; ++s) {
        float bv = -1.0f;
        int   bi = 0x7fffffff;
        float bx = 0.0f, by = 0.0f, bz = 0.0f;

#pragma unroll 4
        for (int j = 0; j < 32; ++j) {
            const size_t o = base4 + (size_t)j * 1024 + tid;
            // gfx1250 prefetch path (global_prefetch_b8): pull next chunks from L2
            if ((j & 7) == 0 && j < 24) {
                __builtin_prefetch((const void*)(xs4 + o + 8 * 1024), 0, 3);
                __builtin_prefetch((const void*)(dd4 + o + 8 * 1024), 0, 3);
            }
            float4 X = xs4[o], Y = ys4[o], Z = zs4[o], D = dd4[o];
            const int p0 = (j * 1024 + tid) * 4;

            float d, nd;
            d  = dist3(X.x - px, Y.x - py, Z.x - pz);
            nd = fminf(D.x, d); D.x = nd;
            if (nd > bv) { bv = nd; bi = p0;     bx = X.x; by = Y.x; bz = Z.x; }

            d  = dist3(X.y - px, Y.y - py, Z.y - pz);
            nd = fminf(D.y, d); D.y = nd;
            if (nd > bv) { bv = nd; bi = p0 + 1; bx = X.y; by = Y.y; bz = Z.y; }

            d  = dist3(X.z - px, Y.z - py, Z.z - pz);
            nd = fminf(D.z, d); D.z = nd;
            if (nd > bv) { bv = nd; bi = p0 + 2; bx = X.z; by = Y.z; bz = Z.z; }

            d  = dist3(X.w - px, Y.w - py, Z.w - pz);
            nd = fminf(D.w, d); D.w = nd;
            if (nd > bv) { bv = nd; bi = p0 + 3; bx = X.w; by = Y.w; bz = Z.w; }

            dd4[o] = D;
        }

        waveReduce5(bv, bi, bx, by, bz);
        if (lane == 0) { sv[wid] = bv; si[wid] = bi; spx[wid] = bx; spy[wid] = by; spz[wid] = bz; }
        __syncthreads();
        if (wid == 0) {
            float v2 = sv[lane];
            int   i2 = si[lane];
            float x2 = spx[lane], y2 = spy[lane], z2 = spz[lane];
            waveReduce5(v2, i2, x2, y2, z2);
            if (lane == 0) {
                idx1[b * NP1 + s] = i2;
                sP[0] = x2; sP[1] = y2; sP[2] = z2;
            }
        }
        __syncthreads();
        px = sP[0]; py = sP[1]; pz = sP[2];
    }
}

// ---- Stages 2-4: FPS on <=1024 gathered points, fully LDS-resident.
//      Gather staged with CDNA5 async global->LDS loads (ASYNCcnt path). ----
__global__ __launch_bounds__(256)
void fps_small(const float* __restrict__ x,
               const int* __restrict__ prev, int prevStride,
               int N, int npoint,
               int* __restrict__ out, int outStride) {
    const int b    = blockIdx.x;
    const int tid  = threadIdx.x;
    const int lane = tid & 31;
    const int wid  = tid >> 5;

    __shared__ float sx[1024], sy[1024], sz[1024], sd[1024];
    __shared__ int   sg[1024];
    __shared__ float sv[8];
    __shared__ int   si[8];
    __shared__ int   sCur;

    const float* xb = x + (size_t)b * N_ * 3;
    const int*   pv = prev + (size_t)b * prevStride;

    for (int i = tid; i < N; i += 256) {
        const int g = pv[i];
        sg[i] = g;
        sd[i] = 1e10f;
        const uint64_t ga  = (uint64_t)(uintptr_t)(xb + (size_t)g * 3);
        const uint64_t gay = ga + 4u;
        const uint64_t gaz = ga + 8u;
        const uint32_t ax = (uint32_t)(uintptr_t)&sx[i];
        const uint32_t ay = (uint32_t)(uintptr_t)&sy[i];
        const uint32_t az = (uint32_t)(uintptr_t)&sz[i];
        asm volatile("global_load_async_to_lds_b32 %0, %1, off" :: "v"(ax), "v"(ga)  : "memory");
        asm volatile("global_load_async_to_lds_b32 %0, %1, off" :: "v"(ay), "v"(gay) : "memory");
        asm volatile("global_load_async_to_lds_b32 %0, %1, off" :: "v"(az), "v"(gaz) : "memory");
    }
    asm volatile("s_wait_asynccnt 0x0" ::: "memory");
    __syncthreads();

    int* ob = out + (size_t)b * outStride;
    if (tid == 0) ob[0] = sg[0];
    int cur = 0;

    for (int s = 1; s < npoint; ++s) {
        const float px = sx[cur], py = sy[cur], pz = sz[cur];
        float bv = -1.0f;
        int   bi = 0x7fffffff;
        for (int i = tid; i < N; i += 256) {
            const float d  = dist3(sx[i] - px, sy[i] - py, sz[i] - pz);
            const float nd = fminf(sd[i], d);
            sd[i] = nd;
            if (nd > bv) { bv = nd; bi = i; }
        }
        waveReduce(bv, bi);
        if (lane == 0) { sv[wid] = bv; si[wid] = bi; }
        __syncthreads();
        if (wid == 0) {
            float v2 = (lane < 8) ? sv[lane] : -1.0f;
            int   i2 = (lane < 8) ? si[lane] : 0x7fffffff;
            waveReduce(v2, i2);
            if (lane == 0) { sCur = i2; ob[s] = sg[i2]; }
        }
        __syncthreads();
        cur = sCur;
    }
}

extern "C" void kernel_launch(void* const* d_in, const int* in_sizes, int n_in,
                              void* d_out, int out_size, void* d_ws, size_t ws_size,
                              hipStream_t stream) {
    const float* x = (const float*)d_in[0];
    int* out = (int*)d_out;

    // Workspace layout: xs | ys | zs | dists | idx1   (~32.06 MB total)
    float* xs = (float*)d_ws;
    float* ys = xs + (size_t)B_ * N_;
    float* zs = ys + (size_t)B_ * N_;
    float* dd = zs + (size_t)B_ * N_;
    int* idx1 = (int*)(dd + (size_t)B_ * N_);

    const size_t total = (size_t)B_ * N_;
    prep_kernel<<<(unsigned)((total + 255) / 256), 256, 0, stream>>>(x, xs, ys, zs, dd);

    fps_big<<<B_, 1024, 0, stream>>>((const float4*)xs, (const float4*)ys,
                                     (const float4*)zs, (float4*)dd,
                                     xs, ys, zs, idx1);

    // Stage 2: 1024 -> 256 (global ids into d_out[0:256))
    fps_small<<<B_, 256, 0, stream>>>(x, idx1, NP1, 1024, 256, out, OUTW);
    // Stage 3: 256 -> 64 (reads stage-2 ids from d_out)
    fps_small<<<B_, 256, 0, stream>>>(x, out, OUTW, 256, 64, out + 256, OUTW);
    // Stage 4: 64 -> 16
    fps_small<<<B_, 256, 0, stream>>>(x, out + 256, OUTW, 64, 16, out + 320, OUTW);
}